// TreeLSTMDP_80229989089609
// MI455X (gfx1250) — compile-verified
//
#include <hip/hip_runtime.h>
#include <math.h>

typedef float v2f __attribute__((ext_vector_type(2)));
typedef float v8f __attribute__((ext_vector_type(8)));

#define HDIM   128
#define NLEAF  8192
#define XS     300
#define NODES  16384
#define BIGF   1e9f
#define EPSF   0.05f
#define LCHUNK 1024

// ---------------- interval helpers ----------------
struct IV { float v, l, u; };
__device__ __forceinline__ IV ivmake(float v, float l, float u){ IV r; r.v=v; r.l=l; r.u=u; return r; }
__device__ __forceinline__ IV ibot(){ return ivmake(0.f, BIGF, -BIGF); }
__device__ __forceinline__ IV imul_(IV a, IV b){
  float c1=a.l*b.l, c2=a.l*b.u, c3=a.u*b.l, c4=a.u*b.u;
  return ivmake(a.v*b.v, fminf(fminf(c1,c2),fminf(c3,c4)), fmaxf(fmaxf(c1,c2),fmaxf(c3,c4)));
}
__device__ __forceinline__ IV iadd_(IV a, IV b){ return ivmake(a.v+b.v, a.l+b.l, a.u+b.u); }
__device__ __forceinline__ IV imerge_(IV a, IV b){
  return ivmake(a.l<=a.u ? a.v : b.v, fminf(a.l,b.l), fmaxf(a.u,b.u));
}
__device__ __forceinline__ IV iaddc(IV a, float b){ return ivmake(a.v+b, a.l+b, a.u+b); }
__device__ __forceinline__ float sigm(float x){ return 1.f/(1.f+expf(-x)); }
__device__ __forceinline__ IV isig(IV a){ return ivmake(sigm(a.v), sigm(a.l), sigm(a.u)); }
__device__ __forceinline__ IV itanh_(IV a){ return ivmake(tanhf(a.v), tanhf(a.l), tanhf(a.u)); }

// node storage: [comp(3)][node(16384)][slot(4)][j(128)]
__device__ __forceinline__ size_t hidx(int comp, int node, int slot, int j){
  return (((size_t)comp*NODES + (size_t)node)*4 + (size_t)slot)*HDIM + (size_t)j;
}

// split tables (delta groups g = d0*2+d1; slot of child = dl0*2+dl1)
__device__ const int QSL9[9] = {0, 0,1, 0,2, 0,1,2,3};
__device__ const int QSR9[9] = {0, 1,0, 2,0, 3,2,1,0};
__device__ const int GQ0T[4] = {0,1,3,5};
__device__ const int GQNT[4] = {1,2,2,4};

// ---------------- weight prep ----------------
// W row-major [Nsrc, K] -> packed B: pairs (B[k][n],B[k+1][n]) at [((kk*2+kh)*Ntot + coloff+n)*2]
__global__ void pack_b(const float* __restrict__ W, float* __restrict__ Bp,
                       int Ntot, int coloff, int Nsrc, int K, int absmode){
  int tid = blockIdx.x*blockDim.x + threadIdx.x;
  int K2 = K >> 1;
  if (tid >= Nsrc*K2) return;
  int nrow = tid / K2, p = tid % K2;
  float w0 = W[(size_t)nrow*K + 2*p];
  float w1 = W[(size_t)nrow*K + 2*p + 1];
  if (absmode){ w0 = fabsf(w0); w1 = fabsf(w1); }
  size_t o = ((size_t)p*Ntot + (size_t)(coloff + nrow))*2;
  Bp[o] = w0; Bp[o+1] = w1;
}

__global__ void absrow_kernel(const float* __restrict__ W_in, float* __restrict__ arow){
  int j = blockIdx.x*blockDim.x + threadIdx.x;
  if (j >= HDIM) return;
  float s = 0.f;
  for (int k = 0; k < XS; ++k) s += fabsf(W_in[(size_t)j*XS + k]);
  arow[j] = EPSF * s;
}

__global__ void gather_emb(const int* __restrict__ x_ids, const float* __restrict__ emb,
                           float* __restrict__ A){
  int tid = blockIdx.x*blockDim.x + threadIdx.x;
  if (tid >= NLEAF*XS) return;
  int m = tid / XS, k = tid % XS;
  A[tid] = emb[(size_t)x_ids[m]*XS + k];
}

// ---------------- fp32 WMMA GEMM: C[M,N] = A[M,K] * B[K,N] ----------------
// grid = (M/16, N/64), block = 32 (one wave). K4 = K/4. B pre-packed by pack_b.
__global__ void wmma_gemm(const float* __restrict__ A, int lda,
                          const float* __restrict__ Bp, int N,
                          float* __restrict__ C, int ldc, int K4){
  int lane = threadIdx.x;
  int m0 = blockIdx.x * 16;
  int n0 = blockIdx.y * 64;
  int ml = lane & 15;
  int kh = lane >> 4;
  const float* arow = A + (size_t)(m0 + ml)*lda + 2*kh;
  v8f acc0 = {}, acc1 = {}, acc2 = {}, acc3 = {};
  for (int kk = 0; kk < K4; ++kk){
    v2f a = *(const v2f*)(arow + 4*kk);
    size_t bb = ((size_t)(2*kk + kh)*N + (size_t)(n0 + ml))*2;
    v2f b0 = *(const v2f*)(Bp + bb);
    v2f b1 = *(const v2f*)(Bp + bb + 32);
    v2f b2 = *(const v2f*)(Bp + bb + 64);
    v2f b3 = *(const v2f*)(Bp + bb + 96);
    acc0 = __builtin_amdgcn_wmma_f32_16x16x4_f32(false, a, false, b0, (short)0, acc0, false, false);
    acc1 = __builtin_amdgcn_wmma_f32_16x16x4_f32(false, a, false, b1, (short)0, acc1, false, false);
    acc2 = __builtin_amdgcn_wmma_f32_16x16x4_f32(false, a, false, b2, (short)0, acc2, false, false);
    acc3 = __builtin_amdgcn_wmma_f32_16x16x4_f32(false, a, false, b3, (short)0, acc3, false, false);
  }
#pragma unroll
  for (int r = 0; r < 8; ++r){
    int row = m0 + r + kh*8;
    float* crow = C + (size_t)row*ldc + (size_t)(n0 + ml);
    crow[0]  = acc0[r];
    crow[16] = acc1[r];
    crow[32] = acc2[r];
    crow[48] = acc3[r];
  }
}

// ---------------- leaf elementwise ----------------
// z = relu(val +/- eps*rowabs + b), emit (v, l+u, u-l)
__global__ void leaf_z(const float* __restrict__ Zin, const float* __restrict__ b_in,
                       const float* __restrict__ arow, float* __restrict__ Zvsd){
  int tid = blockIdx.x*blockDim.x + threadIdx.x;
  if (tid >= NLEAF*HDIM) return;
  int j = tid & (HDIM-1);
  float val = Zin[tid] + b_in[j];
  float a = arow[j];
  float rv = fmaxf(val, 0.f);
  float rl = fmaxf(val - a, 0.f);
  float ru = fmaxf(val + a, 0.f);
  Zvsd[tid] = rv;
  Zvsd[(size_t)NLEAF*HDIM + tid] = rl + ru;
  Zvsd[(size_t)2*NLEAF*HDIM + tid] = ru - rl;
}

__global__ void leaf_apply(const float* __restrict__ IOUL, const float* __restrict__ maskv,
                           const int* __restrict__ unk, const float* __restrict__ b_iou,
                           float* __restrict__ Hb, float* __restrict__ Cb, int* __restrict__ UM){
  int tid = blockIdx.x*blockDim.x + threadIdx.x;
  if (tid >= NLEAF*HDIM) return;
  int m = tid >> 7, j = tid & 127;
  size_t zb = (size_t)m*384;
  size_t cs = (size_t)NLEAF*384;
  float mk = maskv[m];
  IV g[3];
#pragma unroll
  for (int t = 0; t < 3; ++t){
    int col = t*HDIM + j;
    float v = IOUL[zb + col], s = IOUL[cs + zb + col], d = IOUL[2*cs + zb + col];
    IV x = ivmake(v, 0.5f*(s - d), 0.5f*(s + d));
    x.v *= mk; x.l *= mk; x.u *= mk;
    g[t] = iaddc(x, b_iou[col]);
  }
  IV i_ = isig(g[0]);
  IV o_ = isig(g[1]);
  IV u_ = itanh_(g[2]);
  IV c_ = imul_(i_, u_);                 // c_in = 0 at leaves
  IV h_ = imul_(o_, itanh_(c_));
#pragma unroll
  for (int slot = 0; slot < 4; ++slot){
    Hb[hidx(0,m,slot,j)] = h_.v; Hb[hidx(1,m,slot,j)] = h_.l; Hb[hidx(2,m,slot,j)] = h_.u;
    Cb[hidx(0,m,slot,j)] = c_.v; Cb[hidx(1,m,slot,j)] = c_.l; Cb[hidx(2,m,slot,j)] = c_.u;
  }
  if (j == 0) UM[m] = unk[m];
}

// ---------------- tree level: gather h_cat (v, l+u, u-l channels) ----------------
__global__ void gather_hcat(const float* __restrict__ Hb, int cb, int p0,
                            int chunk, int cp, float* __restrict__ Hcat){
  int tid = blockIdx.x*blockDim.x + threadIdx.x;
  int total = 27*cp*256;
  if (tid >= total) return;
  int k = tid & 255;
  int rest = tid >> 8;
  int i = rest % cp; rest /= cp;
  int q = rest % 9;
  int ch = rest / 9;
  size_t o = ((size_t)(ch*9 + q)*cp + i)*256 + k;
  if (i >= chunk){ Hcat[o] = 0.f; return; }
  int p = p0 + i;
  int node = cb + 2*p + (k >= HDIM ? 1 : 0);
  int slot = (k < HDIM) ? QSL9[q] : QSR9[q];
  int j = k & 127;
  float val;
  if (ch == 0) val = Hb[hidx(0,node,slot,j)];
  else {
    float l = Hb[hidx(1,node,slot,j)], u = Hb[hidx(2,node,slot,j)];
    val = (ch == 1) ? (l + u) : (u - l);
  }
  Hcat[o] = val;
}

// ---------------- tree level: merge splits + apply LSTM cell + aux ----------------
__global__ void reduce_apply(const float* __restrict__ Z, int cp,
                             float* __restrict__ Hb, float* __restrict__ Cb,
                             int* __restrict__ UM, int cb, int pb,
                             int p0, int chunk,
                             const float* __restrict__ b_iou,
                             const float* __restrict__ b_f){
  int tid = blockIdx.x*blockDim.x + threadIdx.x;
  if (tid >= chunk*HDIM) return;
  int i = tid >> 7, j = tid & 127;
  int p = p0 + i;
  int n0 = cb + 2*p, n1 = n0 + 1;
  int um0 = UM[n0], um1 = UM[n1];

  for (int g = 0; g < 4; ++g){
    IV pc = ibot(), pi = ibot(), po = ibot(), pu = ibot();
    int q0 = GQ0T[g], qn = GQNT[g];
    for (int s = 0; s < qn; ++s){
      int q = q0 + s;
      size_t r0 = ((size_t)(0*9 + q)*cp + i)*640;
      size_t r1 = ((size_t)(1*9 + q)*cp + i)*640;
      size_t r2 = ((size_t)(2*9 + q)*cp + i)*640;
      // forget gates (U_f outputs are columns 0..255, biased by b_f, then sigmoid)
      float v  = Z[r0 + j], sv = Z[r1 + j], dv = Z[r2 + j];
      IV fl = isig(iaddc(ivmake(v, 0.5f*(sv - dv), 0.5f*(sv + dv)), b_f[j]));
      int c2 = HDIM + j;
      v = Z[r0 + c2]; sv = Z[r1 + c2]; dv = Z[r2 + c2];
      IV fr = isig(iaddc(ivmake(v, 0.5f*(sv - dv), 0.5f*(sv + dv)), b_f[c2]));
      int sl = QSL9[q], sr = QSR9[q];
      IV cl = ivmake(Cb[hidx(0,n0,sl,j)], Cb[hidx(1,n0,sl,j)], Cb[hidx(2,n0,sl,j)]);
      IV cr = ivmake(Cb[hidx(0,n1,sr,j)], Cb[hidx(1,n1,sr,j)], Cb[hidx(2,n1,sr,j)]);
      pc = imerge_(pc, iadd_(imul_(fl, cl), imul_(fr, cr)));
      // iou (U_iou outputs are columns 256..639)
      int ci = 256 + j, co = 384 + j, cu = 512 + j;
      pi = imerge_(pi, ivmake(Z[r0+ci], 0.5f*(Z[r1+ci]-Z[r2+ci]), 0.5f*(Z[r1+ci]+Z[r2+ci])));
      po = imerge_(po, ivmake(Z[r0+co], 0.5f*(Z[r1+co]-Z[r2+co]), 0.5f*(Z[r1+co]+Z[r2+co])));
      pu = imerge_(pu, ivmake(Z[r0+cu], 0.5f*(Z[r1+cu]-Z[r2+cu]), 0.5f*(Z[r1+cu]+Z[r2+cu])));
    }
    int d0 = g >> 1, d1 = g & 1;
    // aux: whole-subtree-consumed cases
    bool condl = (um0 > 0) && (um0 <= d0);
    bool condr = (um1 > 0) && (um1 <= d0);
    int cu0 = um0 < 1 ? 1 : um0; cu0 = cu0 > d0 ? d0 : cu0;
    int cu1 = um1 < 1 ? 1 : um1; cu1 = cu1 > d0 ? d0 : cu1;
    int slL = (d0 - cu0)*2 + d1;   // taken from RIGHT child (index 1)
    int slR = (d0 - cu1)*2 + d1;   // taken from LEFT child (index 0)
    IV ahA = condl ? ivmake(Hb[hidx(0,n1,slL,j)], Hb[hidx(1,n1,slL,j)], Hb[hidx(2,n1,slL,j)]) : ibot();
    IV ahB = condr ? ivmake(Hb[hidx(0,n0,slR,j)], Hb[hidx(1,n0,slR,j)], Hb[hidx(2,n0,slR,j)]) : ibot();
    IV acA = condl ? ivmake(Cb[hidx(0,n1,slL,j)], Cb[hidx(1,n1,slL,j)], Cb[hidx(2,n1,slL,j)]) : ibot();
    IV acB = condr ? ivmake(Cb[hidx(0,n0,slR,j)], Cb[hidx(1,n0,slR,j)], Cb[hidx(2,n0,slR,j)]) : ibot();
    IV auxh = imerge_(ahA, ahB);
    IV auxc = imerge_(acA, acB);
    // apply_dp
    IV i_ = isig(iaddc(pi, b_iou[j]));
    IV o_ = isig(iaddc(po, b_iou[HDIM + j]));
    IV u_ = itanh_(iaddc(pu, b_iou[2*HDIM + j]));
    IV c_ = iadd_(imul_(i_, u_), pc);
    IV h_ = imul_(o_, itanh_(c_));
    h_ = imerge_(h_, auxh);
    c_ = imerge_(c_, auxc);
    int pn = pb + p;
    Hb[hidx(0,pn,g,j)] = h_.v; Hb[hidx(1,pn,g,j)] = h_.l; Hb[hidx(2,pn,g,j)] = h_.u;
    Cb[hidx(0,pn,g,j)] = c_.v; Cb[hidx(1,pn,g,j)] = c_.l; Cb[hidx(2,pn,g,j)] = c_.u;
  }
  if (j == 0) UM[pb + p] = (um0 > 0 && um1 > 0) ? (um0 + um1) : 0;
}

// ---------------- output head ----------------
__global__ void root_kernel(const float* __restrict__ Hb,
                            const float* __restrict__ Wout,
                            const float* __restrict__ bout,
                            float* __restrict__ out){
  int o = threadIdx.x;
  if (o >= 5) return;
  const int node = 16382, slot = 3;   // full-budget slot (D0-1, D1-1, 0)
  float av = bout[o], al = bout[o], au = bout[o];
  for (int j = 0; j < HDIM; ++j){
    float w  = Wout[(size_t)o*HDIM + j];
    float wp = fmaxf(w, 0.f), wn = fminf(w, 0.f);
    float hv = Hb[hidx(0,node,slot,j)];
    float hl = Hb[hidx(1,node,slot,j)];
    float hu = Hb[hidx(2,node,slot,j)];
    av += w*hv; al += wp*hl + wn*hu; au += wp*hu + wn*hl;
  }
  out[o] = av; out[5 + o] = al; out[10 + o] = au;
}

// ---------------- host launch ----------------
extern "C" void kernel_launch(void* const* d_in, const int* in_sizes, int n_in,
                              void* d_out, int out_size, void* d_ws, size_t ws_size,
                              hipStream_t stream){
  (void)in_sizes; (void)n_in; (void)out_size; (void)ws_size;
  const int*   x_ids = (const int*)d_in[0];
  const int*   unk   = (const int*)d_in[1];
  const float* maskv = (const float*)d_in[2];
  const float* emb   = (const float*)d_in[3];
  const float* W_in  = (const float*)d_in[4];
  const float* b_in  = (const float*)d_in[5];
  const float* W_iou = (const float*)d_in[6];
  const float* U_iou = (const float*)d_in[7];
  const float* b_iou = (const float*)d_in[8];
  const float* U_f   = (const float*)d_in[9];
  const float* b_f   = (const float*)d_in[10];
  const float* W_out = (const float*)d_in[11];
  const float* b_out = (const float*)d_in[12];
  float* out = (float*)d_out;

  float* ws = (float*)d_ws;
  size_t off = 0;
  float* HB      = ws + off; off += (size_t)3*NODES*4*HDIM;
  float* CBUF    = ws + off; off += (size_t)3*NODES*4*HDIM;
  int*   UM      = (int*)(ws + off); off += NODES;
  float* B_IN    = ws + off; off += (size_t)XS*HDIM;
  float* AROW    = ws + off; off += HDIM;
  float* B_IOU   = ws + off; off += (size_t)HDIM*384;
  float* B_IOU_A = ws + off; off += (size_t)HDIM*384;
  float* B_CW    = ws + off; off += (size_t)256*640;
  float* B_CW_A  = ws + off; off += (size_t)256*640;
  float* AEMB    = ws + off; off += (size_t)NLEAF*XS;
  float* ZIN     = ws + off; off += (size_t)NLEAF*HDIM;
  float* ZVSD    = ws + off; off += (size_t)3*NLEAF*HDIM;
  float* IOUL    = ws + off; off += (size_t)3*NLEAF*384;
  float* HCAT    = ws + off; off += (size_t)27*LCHUNK*256;
  float* ZLVL    = ws + off; off += (size_t)27*LCHUNK*640;

  // ---- weight prep ----
  {
    int t;
    t = HDIM*(XS/2);  pack_b<<<(t+255)/256,256,0,stream>>>(W_in,  B_IN,    HDIM, 0,   HDIM, XS,  0);
    t = 384*(HDIM/2); pack_b<<<(t+255)/256,256,0,stream>>>(W_iou, B_IOU,   384,  0,   384,  HDIM,0);
                      pack_b<<<(t+255)/256,256,0,stream>>>(W_iou, B_IOU_A, 384,  0,   384,  HDIM,1);
    t = 256*128;      pack_b<<<(t+255)/256,256,0,stream>>>(U_f,   B_CW,    640,  0,   256,  256, 0);
                      pack_b<<<(t+255)/256,256,0,stream>>>(U_f,   B_CW_A,  640,  0,   256,  256, 1);
    t = 384*128;      pack_b<<<(t+255)/256,256,0,stream>>>(U_iou, B_CW,    640,  256, 384,  256, 0);
                      pack_b<<<(t+255)/256,256,0,stream>>>(U_iou, B_CW_A,  640,  256, 384,  256, 1);
    absrow_kernel<<<1,128,0,stream>>>(W_in, AROW);
  }

  // ---- leaf phase ----
  gather_emb<<<(NLEAF*XS+255)/256,256,0,stream>>>(x_ids, emb, AEMB);
  { dim3 g(NLEAF/16, HDIM/64);
    wmma_gemm<<<g,32,0,stream>>>(AEMB, XS, B_IN, HDIM, ZIN, HDIM, XS/4); }
  leaf_z<<<(NLEAF*HDIM+255)/256,256,0,stream>>>(ZIN, b_in, AROW, ZVSD);
  { dim3 g(2*NLEAF/16, 384/64);   // channels v and s share weights W_iou
    wmma_gemm<<<g,32,0,stream>>>(ZVSD, HDIM, B_IOU, 384, IOUL, 384, HDIM/4); }
  { dim3 g(NLEAF/16, 384/64);     // channel d uses |W_iou|
    wmma_gemm<<<g,32,0,stream>>>(ZVSD + (size_t)2*NLEAF*HDIM, HDIM, B_IOU_A, 384,
                                 IOUL + (size_t)2*NLEAF*384, 384, HDIM/4); }
  leaf_apply<<<(NLEAF*HDIM+255)/256,256,0,stream>>>(IOUL, maskv, unk, b_iou, HB, CBUF, UM);

  // ---- tree phase ----
  int cb = 0;
  for (int childn = NLEAF; childn >= 2; childn >>= 1){
    int n = childn >> 1;
    int pb = cb + childn;
    for (int p0 = 0; p0 < n; p0 += LCHUNK){
      int chunk = (n - p0 < LCHUNK) ? (n - p0) : LCHUNK;
      int cp = (chunk + 15) & ~15;
      int tot = 27*cp*256;
      gather_hcat<<<(tot+255)/256,256,0,stream>>>(HB, cb, p0, chunk, cp, HCAT);
      { dim3 g((2*9*cp)/16, 640/64);   // v and s channels against combined [Uf|Uiou]
        wmma_gemm<<<g,32,0,stream>>>(HCAT, 256, B_CW, 640, ZLVL, 640, 256/4); }
      { dim3 g((9*cp)/16, 640/64);     // d channel against |W|
        wmma_gemm<<<g,32,0,stream>>>(HCAT + (size_t)18*cp*256, 256, B_CW_A, 640,
                                     ZLVL + (size_t)18*cp*640, 640, 256/4); }
      int tt = chunk*HDIM;
      reduce_apply<<<(tt+255)/256,256,0,stream>>>(ZLVL, cp, HB, CBUF, UM, cb, pb,
                                                  p0, chunk, b_iou, b_f);
    }
    cb = pb;
  }

  root_kernel<<<1,32,0,stream>>>(HB, W_out, b_out, out);
}